// CRF_18013092839803
// MI455X (gfx1250) — compile-verified
//
#include <hip/hip_runtime.h>
#include <math.h>

// CRF forward on gfx1250: logsumexp-matvec recast as f32 WMMA GEMM vs the
// fixed matrix E = exp(transitions). One wave (32 lanes) owns a 16-batch tile
// and runs both the masked (score) and unmasked (partition) recurrences
// interleaved -> two independent v_wmma chains per step.

typedef __attribute__((ext_vector_type(2))) float v2f;
typedef __attribute__((ext_vector_type(8))) float v8f;

#define NSTATE 48
#define BTILE  16
#define NK     12      // 48 / 4 k-steps per WMMA chain
#define NJ     3       // 48 / 16 column tiles
#define SW     49      // padded LDS row stride (gcd(49,64)=1 -> conflict-free)

#define LOG2E  1.44269504088896340736f
#define LN2    0.69314718055994530942f

// Raw hardware transcendentals (v_exp_f32 / v_log_f32 are base-2).
// Safe here: log inputs are >= min(exp(trans)) >> FLT_MIN (row-max element of
// p is exactly 1), and exp2 denorm-flush only zeroes negligible summands.
__device__ __forceinline__ float fast_exp(float x) {
    return __builtin_amdgcn_exp2f(x * LOG2E);
}
__device__ __forceinline__ float fast_log(float x) {
    return __builtin_amdgcn_logf(x) * LN2;
}

__launch_bounds__(32)
__global__ void crf_fwd_kernel(const float* __restrict__ emis,          // [B,S,T]
                               const unsigned char* __restrict__ mask,  // [B,S] bool
                               const float* __restrict__ start_tr,      // [T]
                               const float* __restrict__ end_tr,        // [T]
                               const float* __restrict__ trans,         // [T,T]
                               float* __restrict__ out,                 // [B]
                               int B, int S)
{
    __shared__ float lds_alpha[2][BTILE * SW];
    __shared__ float lds_p    [2][BTILE * SW];
    __shared__ float lds_m    [2][BTILE];

    const int lane = threadIdx.x;        // 0..31 (wave32)
    const int row  = lane & 15;          // M / N position within tile
    const int half = lane >> 4;          // lane half selects K pair / M+8
    const int b0   = blockIdx.x * BTILE; // batch tile base
    const int jb   = half * 24;          // per-lane 24-state slice for VALU phases

    // ---- loop-invariant B fragments: E = exp(transitions), kept in VGPRs ----
    // Assumed 32-bit 4x16 B layout mirroring the documented A 16x4 layout:
    //   VGPR v, lanes 0-15 : K = 4*kk + v,     N = lane
    //   VGPR v, lanes 16-31: K = 4*kk + v + 2, N = lane-16
    v2f Bfrag[NK][NJ];
    for (int kk = 0; kk < NK; ++kk)
        for (int jt = 0; jt < NJ; ++jt) {
            int n  = jt * 16 + row;
            int k0 = kk * 4 + 2 * half;
            Bfrag[kk][jt][0] = fast_exp(trans[(k0 + 0) * NSTATE + n]);
            Bfrag[kk][jt][1] = fast_exp(trans[(k0 + 1) * NSTATE + n]);
        }

    // ---- alpha0 = start_transitions + emissions[:,0,:] (both passes) ----
    for (int jj = 0; jj < 24; ++jj) {
        int j = jb + jj;
        float a0 = start_tr[j] + emis[((size_t)(b0 + row) * S) * NSTATE + j];
        lds_alpha[0][row * SW + j] = a0;
        lds_alpha[1][row * SW + j] = a0;
    }
    __syncthreads();   // single-wave WG: lowers to S_NOP, acts as LDS fence

    for (int t = 1; t < S; ++t) {
        // prefetch next timestep's emission rows into cache
        if (t + 1 < S)
            __builtin_prefetch(&emis[((size_t)(b0 + row) * S + (t + 1)) * NSTATE + jb], 0, 0);

        // ---- phase 1: per-batch row max m, then p = exp(alpha - m) ----
        for (int q = 0; q < 2; ++q) {
            float av[24];
            float pm = -3.4e38f;
            for (int jj = 0; jj < 24; ++jj) {
                av[jj] = lds_alpha[q][row * SW + jb + jj];
                pm = fmaxf(pm, av[jj]);
            }
            float m = fmaxf(pm, __shfl_xor(pm, 16, 32));   // partner lane ^16
            lds_m[q][row] = m;                             // both halves write same value
            for (int jj = 0; jj < 24; ++jj)
                lds_p[q][row * SW + jb + jj] = fast_exp(av[jj] - m);
        }
        __syncthreads();

        // ---- phase 2: A fragments, ISA 16x4 f32 layout:
        //   lanes 0-15: V0->K=4kk+0, V1->K=4kk+1 ; lanes 16-31: V0->K=4kk+2, V1->K=4kk+3
        v2f Afrag[2][NK];
        for (int q = 0; q < 2; ++q)
            for (int kk = 0; kk < NK; ++kk) {
                int k0 = kk * 4 + 2 * half;
                Afrag[q][kk][0] = lds_p[q][row * SW + k0 + 0];
                Afrag[q][kk][1] = lds_p[q][row * SW + k0 + 1];
            }

        // ---- hoisted per-row (M) values: independent of column tile jt ----
        float m0v[8], m1v[8], mkv[8];
        for (int v = 0; v < 8; ++v) {
            int M = v + half * 8;
            m0v[v] = lds_m[0][M];
            m1v[v] = lds_m[1][M];
            mkv[v] = (float)mask[(size_t)(b0 + M) * S + t];  // 0.0 / 1.0
        }
        __syncthreads();

        // ---- phase 3: WMMA chains (both passes interleaved) + postprocess ----
        for (int jt = 0; jt < NJ; ++jt) {
            v8f a0 = {};   // score-pass accumulator
            v8f a1 = {};   // partition-pass accumulator
            for (int kk = 0; kk < NK; ++kk) {
                a0 = __builtin_amdgcn_wmma_f32_16x16x4_f32(
                        false, Afrag[0][kk], false, Bfrag[kk][jt],
                        (short)0, a0, false, false);
                a1 = __builtin_amdgcn_wmma_f32_16x16x4_f32(
                        false, Afrag[1][kk], false, Bfrag[kk][jt],
                        (short)0, a1, false, false);
            }
            // C/D layout: VGPR v -> (M = v + 8*half, N = jt*16 + row)
            int j = jt * 16 + row;
            for (int v = 0; v < 8; ++v) {
                int M = v + half * 8;
                float e = emis[((size_t)(b0 + M) * S + t) * NSTATE + j];
                // score pass: masked update
                {
                    float na   = m0v[v] + fast_log(a0[v]) + e;
                    float oldv = lds_alpha[0][M * SW + j];
                    lds_alpha[0][M * SW + j] = (mkv[v] != 0.f) ? na : oldv;
                }
                // partition pass: unmasked
                lds_alpha[1][M * SW + j] = m1v[v] + fast_log(a1[v]) + e;
            }
        }
        __syncthreads();
    }

    // ---- finalize: logsumexp(alpha + end_transitions) per batch, per pass ----
    float res[2];
    for (int q = 0; q < 2; ++q) {
        float av[24];
        float pm = -3.4e38f;
        for (int jj = 0; jj < 24; ++jj) {
            int j = jb + jj;
            av[jj] = lds_alpha[q][row * SW + j] + end_tr[j];
            pm = fmaxf(pm, av[jj]);
        }
        float m = fmaxf(pm, __shfl_xor(pm, 16, 32));
        float ps = 0.f;
        for (int jj = 0; jj < 24; ++jj)
            ps += fast_exp(av[jj] - m);
        float s = ps + __shfl_xor(ps, 16, 32);
        res[q] = m + fast_log(s);
    }
    if (lane < 16)
        out[b0 + row] = res[1] - res[0];   // partition - score
}

extern "C" void kernel_launch(void* const* d_in, const int* in_sizes, int n_in,
                              void* d_out, int out_size, void* d_ws, size_t ws_size,
                              hipStream_t stream) {
    const float*         emis  = (const float*)d_in[0];
    const unsigned char* mask  = (const unsigned char*)d_in[1];  // jnp bool -> 1 byte
    const float*         st    = (const float*)d_in[2];
    const float*         en    = (const float*)d_in[3];
    const float*         tr    = (const float*)d_in[4];
    float*               out   = (float*)d_out;

    const int B = 512, S = 1024;
    // one wave per 16-batch tile; 32 independent blocks spread across WGPs
    crf_fwd_kernel<<<dim3(B / BTILE), dim3(32), 0, stream>>>(
        emis, mask, st, en, tr, out, B, S);
}